// GNNEncoder_53558242181892
// MI455X (gfx1250) — compile-verified
//
#include <hip/hip_runtime.h>

typedef __attribute__((ext_vector_type(2))) float v2f;
typedef __attribute__((ext_vector_type(8))) float v8f;

// ---------------- degree / normalization ----------------

__global__ void init_deg_kernel(float* __restrict__ deg, int n) {
  int i = blockIdx.x * blockDim.x + threadIdx.x;
  if (i < n) deg[i] = 1.0f;  // self-loop contributes 1 to every node's degree
}

__global__ void deg_count_kernel(const int* __restrict__ dst, float* __restrict__ deg, int e) {
  int i = blockIdx.x * blockDim.x + threadIdx.x;
  if (i < e) atomicAdd(&deg[dst[i]], 1.0f);
}

__global__ void dinv_kernel(float* __restrict__ deg, int n) {
  int i = blockIdx.x * blockDim.x + threadIdx.x;
  if (i < n) {
    float d = deg[i];
    deg[i] = (d > 0.0f) ? rsqrtf(d) : 0.0f;  // in-place deg -> dinv
  }
}

// ---------------- dense GEMM via native fp32 WMMA ----------------
// H[M,NC] = X[M,K] @ W[K,NC].  One wave computes one 16x16 tile of H.
// blockDim.x = 32 * (NC/16): wave w handles column tile w; blockIdx.x is the M tile.
// Lane mapping per CDNA5 ISA 7.12.2:
//   A (16x4 f32, 2 VGPRs): lanes 0-15 M=lane, VGPR0=K0 ; lanes 16-31 VGPR0=K2.
//   B (4x16 f32, 2 VGPRs): mirrored (row striped across lanes).
//   C/D (16x16 f32, 8 VGPRs): VGPR v = row v (lanes 0-15) / row v+8 (lanes 16-31).
template <int K, int NC>
__global__ __launch_bounds__(32 * (NC / 16)) void gemm_wmma_kernel(
    const float* __restrict__ X, const float* __restrict__ W,
    float* __restrict__ H, int M) {
  const int lane = threadIdx.x & 31;
  const int wave = threadIdx.x >> 5;
  const int m0   = blockIdx.x << 4;
  const int n0   = wave << 4;
  const int l15  = lane & 15;
  const int kh   = (lane >> 4) << 1;  // 0 for lanes 0-15, 2 for lanes 16-31

  int row = m0 + l15;
  if (row >= M) row = M - 1;          // clamp loads; keeps EXEC all-ones for WMMA
  const int col = n0 + l15;

  v8f c = {};
  for (int k0 = 0; k0 < K; k0 += 4) {
    v2f a, b;
    a.x = X[row * K + k0 + kh];
    a.y = X[row * K + k0 + kh + 1];
    b.x = W[(k0 + kh) * NC + col];
    b.y = W[(k0 + kh + 1) * NC + col];
    c = __builtin_amdgcn_wmma_f32_16x16x4_f32(
        /*neg_a=*/false, a, /*neg_b=*/false, b,
        /*c_mod=*/(short)0, c, /*reuse_a=*/false, /*reuse_b=*/false);
  }

  const int rbase = m0 + ((lane >> 4) << 3);
#pragma unroll
  for (int v = 0; v < 8; ++v) {
    int r = rbase + v;
    if (r < M) H[r * NC + col] = c[v];
  }
}

// ---------------- edge gather-scale-scatter ----------------
// One thread per (edge, channel).  C = 1<<CLOG channels.
template <int CLOG>
__global__ void edge_agg_kernel(const int* __restrict__ src, const int* __restrict__ dst,
                                const float* __restrict__ dinv, const float* __restrict__ h,
                                float* __restrict__ agg, int e) {
  unsigned t = blockIdx.x * blockDim.x + threadIdx.x;
  unsigned edge = t >> CLOG;
  if (edge >= (unsigned)e) return;
  unsigned c = t & ((1u << CLOG) - 1u);
  int s = src[edge];
  int d = dst[edge];
  float nrm = dinv[s] * dinv[d];
  atomicAdd(&agg[((size_t)d << CLOG) + c], nrm * h[((size_t)s << CLOG) + c]);
}

// ---------------- self-loop + bias (+ReLU) ----------------
template <int CLOG, bool RELU>
__global__ void finish_kernel(const float* __restrict__ agg, const float* __restrict__ hlin,
                              const float* __restrict__ dinv, const float* __restrict__ bias,
                              float* __restrict__ out, int n) {
  unsigned t = blockIdx.x * blockDim.x + threadIdx.x;
  unsigned i = t >> CLOG;
  if (i >= (unsigned)n) return;
  unsigned c = t & ((1u << CLOG) - 1u);
  float di = dinv[i];
  float v = agg[t] + di * di * hlin[t] + bias[c];
  out[t] = RELU ? fmaxf(v, 0.0f) : v;
}

// ---------------- launch ----------------

extern "C" void kernel_launch(void* const* d_in, const int* in_sizes, int n_in,
                              void* d_out, int out_size, void* d_ws, size_t ws_size,
                              hipStream_t stream) {
  (void)n_in; (void)out_size; (void)ws_size;
  const float* x  = (const float*)d_in[0];
  const int*   ei = (const int*)d_in[1];
  const float* W1 = (const float*)d_in[2];
  const float* b1 = (const float*)d_in[3];
  const float* W2 = (const float*)d_in[4];
  const float* b2 = (const float*)d_in[5];

  const int IN_C = 128, HID = 64, OUT = 32;
  const int n = in_sizes[0] / IN_C;
  const int e = in_sizes[1] / 2;
  const int* srcp = ei;        // edge_index[0]
  const int* dstp = ei + e;    // edge_index[1]

  float* ws   = (float*)d_ws;
  size_t off  = ((size_t)n + 255) & ~(size_t)255;
  float* dinv = ws;                          // n floats (deg, then dinv in place)
  float* h1   = ws + off;                    // n*64
  float* agg1 = h1 + (size_t)n * HID;        // n*64 (post-ReLU layer-1 output, in place)
  float* h2   = agg1 + (size_t)n * HID;      // n*32
  float* outp = (float*)d_out;               // agg2 accumulates straight into d_out

  const int TPB = 256;

  // zero accumulators every call (harness replays the graph; atomics accumulate)
  hipMemsetAsync(agg1, 0, (size_t)n * HID * sizeof(float), stream);
  hipMemsetAsync(d_out, 0, (size_t)n * OUT * sizeof(float), stream);

  // symmetric GCN normalization on A + I
  init_deg_kernel<<<(n + TPB - 1) / TPB, TPB, 0, stream>>>(dinv, n);
  deg_count_kernel<<<(e + TPB - 1) / TPB, TPB, 0, stream>>>(dstp, dinv, e);
  dinv_kernel<<<(n + TPB - 1) / TPB, TPB, 0, stream>>>(dinv, n);

  // ---- layer 1: h1 = x @ W1 ; aggregate ; +dinv^2*h1 + b1 ; ReLU ----
  gemm_wmma_kernel<128, 64><<<(n + 15) / 16, 128, 0, stream>>>(x, W1, h1, n);
  {
    unsigned total = (unsigned)e << 6;
    edge_agg_kernel<6><<<(total + TPB - 1) / TPB, TPB, 0, stream>>>(srcp, dstp, dinv, h1, agg1, e);
  }
  {
    unsigned total = (unsigned)n << 6;
    finish_kernel<6, true><<<(total + TPB - 1) / TPB, TPB, 0, stream>>>(agg1, h1, dinv, b1, agg1, n);
  }

  // ---- layer 2: h2 = relu_out @ W2 ; aggregate into d_out ; +dinv^2*h2 + b2 ----
  gemm_wmma_kernel<64, 32><<<(n + 15) / 16, 64, 0, stream>>>(agg1, W2, h2, n);
  {
    unsigned total = (unsigned)e << 5;
    edge_agg_kernel<5><<<(total + TPB - 1) / TPB, TPB, 0, stream>>>(srcp, dstp, dinv, h2, outp, e);
  }
  {
    unsigned total = (unsigned)n << 5;
    finish_kernel<5, false><<<(total + TPB - 1) / TPB, TPB, 0, stream>>>(outp, h2, dinv, b2, outp, n);
  }
}